// CenterLoss_73246372266128
// MI455X (gfx1250) — compile-verified
//
#include <hip/hip_runtime.h>

#define BATCH   1024
#define FEAT    256
#define NCLASS  100000
#define RSTRIDE 260   // 256 floats + 4 pad -> bank-conflict-free strided v2f reads

typedef __attribute__((ext_vector_type(2))) float v2f;
typedef __attribute__((ext_vector_type(8))) float v8f;

// Low 32 bits of a generic pointer into LDS == byte offset within the
// workgroup's LDS allocation.
__device__ __forceinline__ unsigned lds_off(const void* p) {
    return (unsigned)(size_t)p;
}

// GLOBAL_LOAD_ASYNC_TO_LDS_B128, GVS mode:
//   copies 16 bytes from saddr + vaddr to LDS at vdst, tracked by ASYNCcnt.
__device__ __forceinline__ void async_ld16(unsigned lds_addr, const void* base,
                                           int byte_off) {
    asm volatile("global_load_async_to_lds_b128 %0, %1, %2"
                 :: "v"(lds_addr), "v"(byte_off), "s"(base)
                 : "memory");
}

// One wave (32 lanes) per 16-row tile. Computes clamped per-row squared
// distance d[r] = ||x_r||^2 + ||c_lab(r)||^2 - 2 x_r . c_lab(r) using
// V_WMMA_F32_16X16X4_F32 chains for the cross-dot (diag of X * C^T).
__global__ __launch_bounds__(32)
void center_loss_tile(const float* __restrict__ x,
                      const float* __restrict__ centers,
                      const int* __restrict__ labels,
                      float* __restrict__ row_d) {
    __shared__ float lds[2 * 16 * RSTRIDE];
    float* xs = lds;                    // 16 x 260
    float* cs = lds + 16 * RSTRIDE;     // 16 x 260

    const int lane = threadIdx.x;       // 0..31
    const int r0   = blockIdx.x * 16;

    // ---- Stage tile straight into LDS with async copies (no VGPR bounce) ----
    for (int idx = lane; idx < 16 * (FEAT / 4); idx += 32) {
        const int row = idx >> 6;       // FEAT/4 == 64 float4 per row
        const int c4  = idx & 63;
        async_ld16(lds_off(xs + row * RSTRIDE + c4 * 4), x,
                   ((r0 + row) * FEAT + c4 * 4) * 4);
        const int lbl = labels[r0 + row];
        async_ld16(lds_off(cs + row * RSTRIDE + c4 * 4), centers,
                   (lbl * FEAT + c4 * 4) * 4);
    }
    asm volatile("s_wait_asynccnt 0x0" ::: "memory");
    __syncthreads();

    // ---- WMMA layout: lanes 0-15 hold M=lane,K={k,k+1}; lanes 16-31 hold
    //      M=lane-16,K={k+2,k+3}. B mirrors A (N instead of M). ----
    const int half = lane >> 4;
    const int m    = lane & 15;
    const float* xrow = xs + m * RSTRIDE + 2 * half;
    const float* crow = cs + m * RSTRIDE + 2 * half;

    v8f acc = {};
    float nx = 0.0f, nc = 0.0f;
#pragma unroll 4
    for (int kb = 0; kb < FEAT; kb += 4) {
        v2f a = *(const v2f*)(xrow + kb);
        v2f b = *(const v2f*)(crow + kb);
        nx += a.x * a.x + a.y * a.y;     // partial ||x_m||^2 (this half's K)
        nc += b.x * b.x + b.y * b.y;     // partial ||c_m||^2
        // 8 args: (neg_a, A, neg_b, B, c_mod, C, reuse_a, reuse_b)
        acc = __builtin_amdgcn_wmma_f32_16x16x4_f32(
            false, a, false, b, (short)0, acc, false, false);
    }

    // Combine the two half-wave K-partials of the norms.
    nx += __shfl_xor(nx, 16, 32);
    nc += __shfl_xor(nc, 16, 32);

    // ---- Spill acc (16x16) to LDS, pull the diagonal ----
    __syncthreads();
    float* out = lds;                   // reuse as 16 x 17
#pragma unroll
    for (int v = 0; v < 8; ++v) {
        // C/D layout: VGPR v -> M = v + 8*half, N = lane&15
        out[(v + 8 * half) * 17 + m] = acc[v];
    }
    __syncthreads();

    if (lane < 16) {
        const float dot = out[lane * 17 + lane];
        float d = nx + nc - 2.0f * dot;
        d = fminf(fmaxf(d, 1e-12f), 1e12f);
        row_d[r0 + lane] = d;
    }
}

// Deterministic single-block tree reduction of the 1024 per-row distances.
__global__ __launch_bounds__(256)
void center_loss_reduce(const float* __restrict__ row_d,
                        float* __restrict__ out) {
    __shared__ float s[256];
    const int t = threadIdx.x;
    float v = row_d[t] + row_d[t + 256] + row_d[t + 512] + row_d[t + 768];
    s[t] = v;
    __syncthreads();
    for (int off = 128; off > 0; off >>= 1) {
        if (t < off) s[t] += s[t + off];
        __syncthreads();
    }
    if (t == 0) {
        out[0] = s[0] * (1.0f / BATCH) + (float)((NCLASS - 1) * 1e-12);
    }
}

extern "C" void kernel_launch(void* const* d_in, const int* in_sizes, int n_in,
                              void* d_out, int out_size, void* d_ws, size_t ws_size,
                              hipStream_t stream) {
    (void)in_sizes; (void)n_in; (void)out_size; (void)ws_size;
    const float* x       = (const float*)d_in[0];
    const float* centers = (const float*)d_in[1];
    const int*   labels  = (const int*)d_in[2];
    float* row_d = (float*)d_ws;            // 1024 floats of scratch

    center_loss_tile<<<BATCH / 16, 32, 0, stream>>>(x, centers, labels, row_d);
    center_loss_reduce<<<1, 256, 0, stream>>>(row_d, (float*)d_out);
}